// rgcnModel_29540785062551
// MI455X (gfx1250) — compile-verified
//
#include <hip/hip_runtime.h>

// ---------------------------------------------------------------------------
// RGCN forward for MI455X (gfx1250, wave32, WMMA).
//   h = x@W1+b1
//   2x: out = h@root + bias_c;  hr[r] = h@weight[r];  out[dst]+=hr[et][src];
//       h = relu(bn(out))
//   t = h@W2+b2;  y = sigmoid(t@Wo+bo)
// GEMMs: V_WMMA_F32_16X16X4_F32 (exact fp32), 64x128 block tiles, B staged in
// LDS via GLOBAL_LOAD_ASYNC_TO_LDS_B128 (ASYNCcnt). Edge scatter:
// global_atomic_add_f32 into an L2-resident 25.6MB buffer; 205MB hr gather is
// non-temporal so out/h stay resident in the 192MB L2.
// ---------------------------------------------------------------------------

typedef float v2f __attribute__((ext_vector_type(2)));
typedef float v4f __attribute__((ext_vector_type(4)));
typedef float v8f __attribute__((ext_vector_type(8)));

#define HD     128
#define NNODES 50000
#define RREL   8
#define EEDGES 800000
#define BNEPS  1e-5f

#define USE_ASYNC_LDS 1   // flip to 0 if assembler rejects async mnemonic

// C[M,128] = A[M,128] @ B[128,128] (+bias). blockIdx.y selects B/C (batched
// relation GEMMs). 128 threads = 4 waves. Block tile: 64 rows x 128 cols.
// Wave w: cols [32w, 32w+32), all 64 rows (4 row-groups of 16), 8 v8f accs.
__global__ __launch_bounds__(128) void gemm128_wmma(
    const float* __restrict__ A, const float* __restrict__ Bbase,
    const float* __restrict__ bias, float* __restrict__ Cbase,
    long strideB, long strideC, int M)
{
  const float* B = Bbase + (long)blockIdx.y * strideB;
  float*       C = Cbase + (long)blockIdx.y * strideC;

  __shared__ float Bs[HD * HD];                 // 64 KB of 320 KB/WGP

#if USE_ASYNC_LDS
  // Async DMA fill of Bs: each lane moves 16B per issue, 32 issues/thread.
  {
    const char* gsrc = (const char*)B;
    #pragma unroll 4
    for (int i = threadIdx.x * 16; i < HD * HD * 4; i += 128 * 16) {
      unsigned           lds = (unsigned)(uintptr_t)((char*)Bs + i);
      unsigned long long ga  = (unsigned long long)(gsrc + i);
      asm volatile("global_load_async_to_lds_b128 %0, %1, off"
                   :: "v"(lds), "v"(ga) : "memory");
    }
    asm volatile("s_wait_asynccnt 0" ::: "memory");
  }
#else
  for (int i = threadIdx.x; i < HD * HD; i += 128) Bs[i] = B[i];
#endif
  __syncthreads();

  const int lane  = threadIdx.x & 31;
  const int wave  = threadIdx.x >> 5;
  const int m0    = blockIdx.x * 64;
  const int n0    = wave * 32;
  const int khalf = (lane >> 4) * 2;            // lanes 0-15: K+0/1, 16-31: K+2/3
  const int ncol  = lane & 15;

  // Clamp A-row indices so every lane loads in-bounds (EXEC must stay all-1s
  // for WMMA); stores are predicated per 16-row group instead.
  const float* Arow[4];
  #pragma unroll
  for (int g = 0; g < 4; ++g) {
    int r = m0 + g * 16 + (lane & 15);
    if (r > M - 1) r = M - 1;
    Arow[g] = A + (long)r * HD;
  }

  v8f acc[4][2];
  #pragma unroll
  for (int g = 0; g < 4; ++g) { acc[g][0] = (v8f){}; acc[g][1] = (v8f){}; }

  #pragma unroll 4
  for (int k = 0; k < HD; k += 4) {
    const int kk = k + khalf;
    v2f b0, b1;
    b0.x = Bs[kk * HD + n0 + ncol];
    b0.y = Bs[(kk + 1) * HD + n0 + ncol];
    b1.x = Bs[kk * HD + n0 + 16 + ncol];
    b1.y = Bs[(kk + 1) * HD + n0 + 16 + ncol];
    #pragma unroll
    for (int g = 0; g < 4; ++g) {
      v2f a = *(const v2f*)(Arow[g] + kk);      // A[m][kk], A[m][kk+1]
      acc[g][0] = __builtin_amdgcn_wmma_f32_16x16x4_f32(false, a, false, b0,
                                                (short)0, acc[g][0], false, false);
      acc[g][1] = __builtin_amdgcn_wmma_f32_16x16x4_f32(false, a, false, b1,
                                                (short)0, acc[g][1], false, false);
    }
  }

  // C/D layout: vgpr i holds row base + i + 8*(lane>=16), col n0 + (lane&15)
  const float bv0 = bias ? bias[n0 + ncol]      : 0.f;
  const float bv1 = bias ? bias[n0 + 16 + ncol] : 0.f;
  #pragma unroll
  for (int g = 0; g < 4; ++g) {
    if (m0 + g * 16 >= M) break;                // 50000 % 16 == 0: whole group
    const int mbase = m0 + g * 16 + (lane >> 4) * 8;
    #pragma unroll
    for (int i = 0; i < 8; ++i) {
      C[(long)(mbase + i) * HD + n0 + ncol]      = acc[g][0][i] + bv0;
      C[(long)(mbase + i) * HD + n0 + 16 + ncol] = acc[g][1][i] + bv1;
    }
  }
}

// One wave per edge: out[dst,:] += hr[etype, src, :]; lane handles 4 floats.
// hr (205MB) is streamed non-temporally; atomics land in a 25.6MB
// L2-resident buffer and execute at the L2 atomic units.
__global__ __launch_bounds__(256) void edge_scatter(
    const int* __restrict__ ei, const int* __restrict__ et,
    const float* __restrict__ hr, float* __restrict__ out)
{
  const long tid  = (long)blockIdx.x * 256 + threadIdx.x;
  const int  e    = (int)(tid >> 5);
  const int  lane = (int)(tid & 31);
  if (e >= EEDGES) return;
  const int src = ei[e];
  const int dst = ei[EEDGES + e];
  const int r   = et[e];
  const int c   = lane * 4;
  const v4f v = __builtin_nontemporal_load(
      (const v4f*)(hr + ((long)r * NNODES + src) * HD + c));
  float* o = out + (long)dst * HD + c;
  unsafeAtomicAdd(o + 0, v.x);
  unsafeAtomicAdd(o + 1, v.y);
  unsafeAtomicAdd(o + 2, v.z);
  unsafeAtomicAdd(o + 3, v.w);
}

__global__ void zero_f32(float* p, int n) {
  int i = blockIdx.x * blockDim.x + threadIdx.x;
  if (i < n) p[i] = 0.f;
}

// Per-column sum / sumsq partials (coalesced: thread c reads column c).
__global__ __launch_bounds__(128) void bn_stats(
    const float* __restrict__ x, float* __restrict__ stats)
{
  const int c    = threadIdx.x;
  const int row0 = blockIdx.x * 250;            // 200 blocks * 250 rows = N
  float s = 0.f, q = 0.f;
  for (int i = 0; i < 250; ++i) {
    const float v = x[(long)(row0 + i) * HD + c];
    s += v; q += v * v;
  }
  unsafeAtomicAdd(&stats[c], s);
  unsafeAtomicAdd(&stats[HD + c], q);
}

__global__ void bn_finalize(const float* __restrict__ stats,
                            const float* __restrict__ gamma,
                            const float* __restrict__ beta,
                            float* __restrict__ sc)
{
  const int c = threadIdx.x;
  const float inv_n = 1.f / (float)NNODES;
  const float mu    = stats[c] * inv_n;
  const float var   = stats[HD + c] * inv_n - mu * mu;   // biased var
  const float scale = gamma[c] * rsqrtf(var + BNEPS);
  sc[c]      = scale;
  sc[HD + c] = beta[c] - mu * scale;
}

__global__ __launch_bounds__(256) void bn_apply_relu(
    const float* __restrict__ x, const float* __restrict__ sc,
    float* __restrict__ h)
{
  const long i = (long)blockIdx.x * 256 + threadIdx.x;
  const int  c = (int)(i & (HD - 1));
  const float v = x[i] * sc[c] + sc[HD + c];
  h[i] = v > 0.f ? v : 0.f;
}

// One wave per row: y = sigmoid(dot(t[row,:], Wo) + bo)
__global__ __launch_bounds__(256) void head_sigmoid(
    const float* __restrict__ t, const float* __restrict__ Wo,
    const float* __restrict__ bo, float* __restrict__ y)
{
  const long tid  = (long)blockIdx.x * 256 + threadIdx.x;
  const int  row  = (int)(tid >> 5);
  const int  lane = (int)(tid & 31);
  if (row >= NNODES) return;
  const float4 a = *(const float4*)(t + (long)row * HD + lane * 4);
  const float4 w = *(const float4*)(Wo + lane * 4);
  float s = a.x * w.x + a.y * w.y + a.z * w.z + a.w * w.w;
  #pragma unroll
  for (int m = 16; m >= 1; m >>= 1) s += __shfl_xor(s, m, 32);
  if (lane == 0) y[row] = 1.f / (1.f + __expf(-(s + bo[0])));
}

extern "C" void kernel_launch(void* const* d_in, const int* in_sizes, int n_in,
                              void* d_out, int out_size, void* d_ws, size_t ws_size,
                              hipStream_t stream) {
  const float* x      = (const float*)d_in[0];
  const int*   ei     = (const int*)  d_in[1];   // [2,E]
  const int*   et     = (const int*)  d_in[2];   // [E]
  const float* W1     = (const float*)d_in[3];
  const float* b1     = (const float*)d_in[4];
  const float* weight = (const float*)d_in[5];   // [R,H,H]
  const float* root   = (const float*)d_in[6];
  const float* bias_c = (const float*)d_in[7];
  const float* gamma  = (const float*)d_in[8];
  const float* beta   = (const float*)d_in[9];
  const float* W2     = (const float*)d_in[10];
  const float* b2     = (const float*)d_in[11];
  const float* Wo     = (const float*)d_in[12];
  const float* bo     = (const float*)d_in[13];
  float* y = (float*)d_out;

  // workspace layout
  float* h     = (float*)d_ws;                          // N*H
  float* tmp   = h   + (size_t)NNODES * HD;             // N*H
  float* hr    = tmp + (size_t)NNODES * HD;             // R*N*H (204.8 MB)
  float* stats = hr  + (size_t)RREL * NNODES * HD;      // 2H sums
  float* sc    = stats + 2 * HD;                        // 2H scale/shift

  const int  rowBlocks = (NNODES + 63) / 64;            // 782 (last: 16 rows)
  const dim3 gemmGrid1(rowBlocks, 1);
  const dim3 gemmGridR(rowBlocks, RREL);
  const int  edgeBlocks = (EEDGES * 32) / 256;          // 100000
  const int  bnApplyBlocks = (NNODES * HD) / 256;       // 25000
  const int  headBlocks = (NNODES * 32) / 256;          // 6250

  // h = x @ W1 + b1
  gemm128_wmma<<<gemmGrid1, 128, 0, stream>>>(x, W1, b1, h, 0, 0, NNODES);

  for (int layer = 0; layer < 2; ++layer) {
    // out := h @ root + bias_c (scatter accumulates on top)
    gemm128_wmma<<<gemmGrid1, 128, 0, stream>>>(h, root, bias_c, tmp, 0, 0, NNODES);
    // hr[r] = h @ weight[r], r = 0..7 (batched over blockIdx.y)
    gemm128_wmma<<<gemmGridR, 128, 0, stream>>>(
        h, weight, nullptr, hr, (long)HD * HD, (long)NNODES * HD, NNODES);
    zero_f32<<<1, 256, 0, stream>>>(stats, 2 * HD);
    edge_scatter<<<edgeBlocks, 256, 0, stream>>>(ei, et, hr, tmp);
    bn_stats<<<200, 128, 0, stream>>>(tmp, stats);
    bn_finalize<<<1, 128, 0, stream>>>(stats, gamma, beta, sc);
    bn_apply_relu<<<bnApplyBlocks, 256, 0, stream>>>(tmp, sc, h);
  }

  // t = h @ W2 + b2 ; y = sigmoid(t @ Wo + bo)
  gemm128_wmma<<<gemmGrid1, 128, 0, stream>>>(h, W2, b2, tmp, 0, 0, NNODES);
  head_sigmoid<<<headBlocks, 256, 0, stream>>>(tmp, Wo, bo, y);
}